// RotatedIoULoss_13219909337487
// MI455X (gfx1250) — compile-verified
//
#include <hip/hip_runtime.h>
#include <math.h>

typedef __attribute__((ext_vector_type(2))) float v2f;
typedef __attribute__((ext_vector_type(8))) float v8f;

// Wave32 sum via V_WMMA_F32_16X16X4_F32.
// A (16x4 f32, 2 VGPRs/lane) carries each lane's partial in a.x (a.y = 0).
// B = all ones (4x16, layout-independent) -> D[m][n] = rowsum(A)[m].
// Column n of D lives in lane n (M=0..7) and lane n+16 (M=8..15); summing a
// lane's 8 D regs and xor-shuffling lane^16 yields the full wave sum.
// Requires EXEC == all ones at the call site.
__device__ __forceinline__ float wave_sum_wmma(float t) {
  v2f a; a.x = t;    a.y = 0.0f;
  v2f b; b.x = 1.0f; b.y = 1.0f;
  v8f c = {0.f, 0.f, 0.f, 0.f, 0.f, 0.f, 0.f, 0.f};
  v8f d = __builtin_amdgcn_wmma_f32_16x16x4_f32(
      /*neg_a=*/false, a, /*neg_b=*/false, b,
      /*c_mod=*/(short)0, c, /*reuse_a=*/false, /*reuse_b=*/false);
  float s = d[0] + d[1] + d[2] + d[3] + d[4] + d[5] + d[6] + d[7];
  s += __shfl_xor(s, 16, 32);
  return s;  // total in every lane
}

// Branchless compare-exchange (ascending by key).  1 v_cmp + 6 v_cndmask.
__device__ __forceinline__ void ce(float& k0, float& k1,
                                   float& x0, float& x1,
                                   float& y0, float& y1) {
  bool sw = k0 > k1;
  float tk = sw ? k1 : k0; k1 = sw ? k0 : k1; k0 = tk;
  float tx = sw ? x1 : x0; x1 = sw ? x0 : x1; x0 = tx;
  float ty = sw ? y1 : y0; y1 = sw ? y0 : y1; y0 = ty;
}

// Diamond pseudo-angle: monotone with atan2 counterclockwise, range [0,4),
// branch cut at +x instead of -x (shoelace area is rotation-invariant).
__device__ __forceinline__ float pseudo_angle(float dx, float dy) {
  float ad = fabsf(dx) + fabsf(dy);
  float sf = (ad > 0.f) ? ad : 1.f;
  float a  = dx / sf;
  return (dy >= 0.f) ? (1.f - a) : (3.f + a);
}

__device__ float rotated_iou_loss_elem(const float* __restrict__ pred,
                                       const float* __restrict__ target,
                                       const float* __restrict__ weight,
                                       int i) {
  const float* P = pred   + (size_t)i * 5;
  const float* T = target + (size_t)i * 5;
  float ax = P[0], ay = P[1], aw = P[2], ah = P[3], aang = P[4];
  float bx = T[0], by = T[1], bw = T[2], bh = T[3], bang = T[4];
  float ca = cosf(aang), sa = sinf(aang);
  float cb = cosf(bang), sb = sinf(bang);

  const float sdx[4] = {-0.5f,  0.5f, 0.5f, -0.5f};
  const float sdy[4] = {-0.5f, -0.5f, 0.5f,  0.5f};
  float Axc[4], Ayc[4], Bxc[4], Byc[4];
#pragma unroll
  for (int k = 0; k < 4; ++k) {
    float dx = sdx[k] * aw, dy = sdy[k] * ah;
    Axc[k] = ax + dx * ca - dy * sa;
    Ayc[k] = ay + dx * sa + dy * ca;
    float ex = sdx[k] * bw, ey = sdy[k] * bh;
    Bxc[k] = bx + ex * cb - ey * sb;
    Byc[k] = by + ex * sb + ey * cb;
  }

  const float tol = 1e-5f;
  float px[24], py[24], key[24];
  int   msk[24];

  // corners of A inside B / corners of B inside A (bitwise &, no branches)
#pragma unroll
  for (int k = 0; k < 4; ++k) {
    float dx = Axc[k] - bx, dy = Ayc[k] - by;
    float lx =  dx * cb + dy * sb;
    float ly = -dx * sb + dy * cb;
    msk[k] = (int)(fabsf(lx) <= bw * 0.5f + tol) & (int)(fabsf(ly) <= bh * 0.5f + tol);
    px[k] = Axc[k]; py[k] = Ayc[k];
  }
#pragma unroll
  for (int k = 0; k < 4; ++k) {
    float dx = Bxc[k] - ax, dy = Byc[k] - ay;
    float lx =  dx * ca + dy * sa;
    float ly = -dx * sa + dy * ca;
    msk[4 + k] = (int)(fabsf(lx) <= aw * 0.5f + tol) & (int)(fabsf(ly) <= ah * 0.5f + tol);
    px[4 + k] = Bxc[k]; py[4 + k] = Byc[k];
  }
  // 4x4 edge intersections (A-edge major, matching reference reshape)
#pragma unroll
  for (int e = 0; e < 4; ++e) {
    float p0x = Axc[e], p0y = Ayc[e];
    float rx = Axc[(e + 1) & 3] - p0x, ry = Ayc[(e + 1) & 3] - p0y;
#pragma unroll
    for (int f = 0; f < 4; ++f) {
      float q0x = Bxc[f], q0y = Byc[f];
      float sx = Bxc[(f + 1) & 3] - q0x, sy = Byc[(f + 1) & 3] - q0y;
      float den = rx * sy - ry * sx;
      int   ok  = (int)(fabsf(den) > 1e-9f);
      float safe = ok ? den : 1.0f;
      float qpx = q0x - p0x, qpy = q0y - p0y;
      float t = (qpx * sy - qpy * sx) / safe;
      float u = (qpx * ry - qpy * rx) / safe;
      int vld = ok & (int)(t >= 0.f) & (int)(t <= 1.f)
                   & (int)(u >= 0.f) & (int)(u <= 1.f);
      int k = 8 + e * 4 + f;
      msk[k] = vld;
      px[k] = vld ? (p0x + t * rx) : 0.f;
      py[k] = vld ? (p0y + t * ry) : 0.f;
    }
  }

  // centroid of masked points
  int num = 0; float sxs = 0.f, sys = 0.f;
#pragma unroll
  for (int k = 0; k < 24; ++k) {
    num += msk[k];
    sxs += msk[k] ? px[k] : 0.f;
    sys += msk[k] ? py[k] : 0.f;
  }
  float inv  = 1.0f / (float)(num < 1 ? 1 : num);
  float cenx = sxs * inv, ceny = sys * inv;

#pragma unroll
  for (int k = 0; k < 24; ++k) {
    key[k] = msk[k] ? pseudo_angle(px[k] - cenx, py[k] - ceny) : 1e9f;
  }

  // Batcher odd-even mergesort, n = 24 (valid sub-network of the n=32 net).
  // All bounds compile-time constant -> fully unrolled, arrays stay in VGPRs.
#pragma unroll
  for (int p = 1; p < 24; p <<= 1) {
#pragma unroll
    for (int q = p; q >= 1; q >>= 1) {
#pragma unroll
      for (int j = q % p; j + q < 24; j += 2 * q) {
#pragma unroll
        for (int t = 0; t < q; ++t) {
          if (t + j + q < 24) {
            if ((t + j) / (2 * p) == (t + j + q) / (2 * p)) {
              ce(key[t + j], key[t + j + q],
                 px[t + j],  px[t + j + q],
                 py[t + j],  py[t + j + q]);
            }
          }
        }
      }
    }
  }

  // v = (masked ? pt : first) - centroid   (mask after sort: key < 5)
  float fx = px[0], fy = py[0];
#pragma unroll
  for (int k = 0; k < 24; ++k) {
    bool m = key[k] < 5.0f;
    px[k] = (m ? px[k] : fx) - cenx;
    py[k] = (m ? py[k] : fy) - ceny;
  }
  float cr = 0.f;
#pragma unroll
  for (int k = 0; k < 24; ++k) {
    int k2 = (k + 1) % 24;
    cr += px[k] * py[k2] - py[k] * px[k2];
  }
  float inter = (num >= 3) ? 0.5f * fabsf(cr) : 0.f;

  float areaA = aw * ah, areaB = bw * bh;
  float uni = areaA + areaB - inter;
  float iou = inter / fmaxf(uni, 1e-9f);
  iou = fmaxf(iou, 1e-6f);
  return -logf(iou) * weight[i];
}

__global__ void __launch_bounds__(256)
k_iou_partial(const float* __restrict__ pred,
              const float* __restrict__ target,
              const float* __restrict__ weight,
              float* __restrict__ partial, int N, int iters) {
  int tid0   = blockIdx.x * blockDim.x + threadIdx.x;
  int stride = gridDim.x * blockDim.x;
  float acc = 0.f;
  // uniform trip count: all lanes converged at the WMMA below
  for (int k = 0; k < iters; ++k) {
    int  i     = tid0 + k * stride;
    bool valid = i < N;
    int  j     = valid ? i : 0;
    float l = rotated_iou_loss_elem(pred, target, weight, j);
    acc += valid ? l : 0.f;
  }
  float wsum = wave_sum_wmma(acc);
  __shared__ float sm[8];
  int lane = threadIdx.x & 31, wid = threadIdx.x >> 5;
  if (lane == 0) sm[wid] = wsum;
  __syncthreads();
  if (threadIdx.x == 0) {
    float s = 0.f;
    int nw = (int)(blockDim.x + 31) >> 5;
    for (int t = 0; t < nw; ++t) s += sm[t];
    partial[blockIdx.x] = s;
  }
}

__global__ void __launch_bounds__(256)
k_iou_final(const float* __restrict__ partial, int nPart,
            float* __restrict__ out, float scale) {
  float s = 0.f;
  int iters = (nPart + (int)blockDim.x - 1) / (int)blockDim.x;
  for (int k = 0; k < iters; ++k) {
    int t  = (int)threadIdx.x + k * (int)blockDim.x;
    int tt = (t < nPart) ? t : 0;
    float v = partial[tt];
    s += (t < nPart) ? v : 0.f;
  }
  float wsum = wave_sum_wmma(s);
  __shared__ float sm[8];
  int lane = threadIdx.x & 31, wid = threadIdx.x >> 5;
  if (lane == 0) sm[wid] = wsum;
  __syncthreads();
  if (threadIdx.x == 0) {
    float tot = 0.f;
    int nw = (int)(blockDim.x + 31) >> 5;
    for (int t = 0; t < nw; ++t) tot += sm[t];
    out[0] = tot * scale;  // mean * LOSS_WEIGHT(=1)
  }
}

extern "C" void kernel_launch(void* const* d_in, const int* in_sizes, int n_in,
                              void* d_out, int out_size, void* d_ws, size_t ws_size,
                              hipStream_t stream) {
  const float* pred   = (const float*)d_in[0];
  const float* target = (const float*)d_in[1];
  const float* weight = (const float*)d_in[2];
  float* out = (float*)d_out;
  float* partial = (float*)d_ws;

  int N = in_sizes[2];  // weight has N elements
  if (N <= 0) return;

  const int block = 256;
  int grid = (N + block - 1) / block;
  int maxPart = (int)(ws_size / sizeof(float));
  if (maxPart < 1) maxPart = 1;
  if (maxPart > 4096) maxPart = 4096;
  if (grid > maxPart) grid = maxPart;
  if (grid < 1) grid = 1;
  int stride = grid * block;
  int iters = (N + stride - 1) / stride;

  k_iou_partial<<<grid, block, 0, stream>>>(pred, target, weight, partial, N, iters);
  k_iou_final<<<1, block, 0, stream>>>(partial, grid, out, 1.0f / (float)N);
}